// EMAVectorQuantizer_32074815767047
// MI455X (gfx1250) — compile-verified
//
#include <hip/hip_runtime.h>
#include <hip/hip_bf16.h>
#include <math.h>

typedef __attribute__((ext_vector_type(2))) float v2f;
typedef __attribute__((ext_vector_type(8))) float v8f;
typedef __attribute__((ext_vector_type(4))) int   v4i;
typedef __attribute__((address_space(3))) float   lfloat;  // LDS pointer (32-bit)
typedef __attribute__((address_space(1))) v4i     g4i;
typedef __attribute__((address_space(3))) v4i     l4i;

#define N_E     8192
#define E_DIM   256
#define N_ROWS  16384          // 4 * 16*16*16
#define LHW     4096           // 16*16*16
#define BETA    0.25f
#define TILE_N  16
#define N_TILES (N_E / TILE_N) // 512
#define LDS_STRIDE 260         // 256 + 4 pad -> conflict-free b64 LDS reads

// ws layout (bytes)
#define WS_COUNTS_OFF 16
#define WS_IDX_OFF    (16 + N_E * 4)
#define WS_WNORM_OFF  (16 + N_E * 4 + N_ROWS * 4)

// ---- CDNA5 async global->LDS copy (ASYNCcnt-tracked) ----
__device__ __forceinline__ void async_copy16(const float* g, lfloat* l) {
#if __has_builtin(__builtin_amdgcn_global_load_async_to_lds_b128)
  __builtin_amdgcn_global_load_async_to_lds_b128((g4i*)g, (l4i*)l, 0, 0);
#else
  unsigned int loff = (unsigned int)(unsigned long long)l;
  asm volatile("global_load_async_to_lds_b128 %0, %1, off"
               :: "v"(loff), "v"(g) : "memory");
#endif
}

__device__ __forceinline__ void wait_async_le4() {
#if __has_builtin(__builtin_amdgcn_s_wait_asynccnt)
  __builtin_amdgcn_s_wait_asynccnt(4);
#else
  asm volatile("s_wait_asynccnt 0x4" ::: "memory");
#endif
}

// ---------------- init: zero accumulators ----------------
__global__ void vq_init(float* __restrict__ sumsq, int* __restrict__ counts) {
  int t = blockIdx.x * blockDim.x + threadIdx.x;
  if (t == 0) *sumsq = 0.0f;
  for (int i = t; i < N_E; i += gridDim.x * blockDim.x) counts[i] = 0;
}

// ---------------- ||w||^2 per code: one wave per row ----------------
__global__ void vq_wnorm(const float* __restrict__ w, float* __restrict__ wnorm) {
  const int wave = threadIdx.x >> 5;
  const int lane = threadIdx.x & 31;
  const int row  = blockIdx.x * 8 + wave;
  const float* wr = w + (size_t)row * E_DIM;
  float s = 0.0f;
#pragma unroll
  for (int i = 0; i < 8; ++i) { float v = wr[lane + i * 32]; s += v * v; }
#pragma unroll
  for (int m = 16; m >= 1; m >>= 1) s += __shfl_xor(s, m);
  if (lane == 0) wnorm[row] = s;
}

// ---------------- main: fp32 WMMA distance GEMM + fused argmin ----------------
// block = 256 threads = 8 waves; each wave owns a 16-row tile of zf.
// Double-buffered LDS weight tiles filled by async global->LDS DMA; loop
// unrolled x2 so buffer roles are compile-time (no v_movrels, no dyn indexing).
__global__ __launch_bounds__(256)
void vq_argmin(const float* __restrict__ z, const float* __restrict__ w,
               const float* __restrict__ wnorm, int* __restrict__ idxI,
               float* __restrict__ idxF, int* __restrict__ counts) {
  __shared__ float ldsW[2][TILE_N * LDS_STRIDE];

  const int t    = threadIdx.x;
  const int wave = t >> 5;
  const int lane = t & 31;
  const int col  = lane & 15;    // output column / A row within tile
  const int half = lane >> 4;    // 0: K+0/K+1, 1: K+2/K+3 (A & B frag layout)
  const int koff = half * 2;

  const int rowBase = blockIdx.x * 128 + wave * 16;
  const int m   = rowBase + col;      // this lane's zf row
  const int b   = m >> 12;            // batch
  const int lhw = m & (LHW - 1);      // spatial index
  const float* zb = z + (size_t)b * (E_DIM * LHW) + lhw;

  // Staging bases: slot i is +i*4*E_DIM (global) / +i*4*LDS_STRIDE (LDS).
  const int code0 = t >> 6;                 // 0..3
  const int d4    = (t & 63) << 2;
  const float* g0 = w + (size_t)code0 * E_DIM + d4;
  lfloat* l0 = (lfloat*)&ldsW[0][code0 * LDS_STRIDE + d4];  // AS3 cast once
  lfloat* l1 = (lfloat*)&ldsW[1][code0 * LDS_STRIDE + d4];

  // A-stationary: 16x256 fp32 tile as 64 K-fragments (128 VGPRs), loaded once.
  v2f A[64];
#pragma unroll
  for (int k = 0; k < 64; ++k) {
    const int K0 = 4 * k + koff;
    A[k].x = zb[(size_t)K0 * LHW];
    A[k].y = zb[(size_t)(K0 + 1) * LHW];
  }

  float minv[8];
  int   mini[8];
#pragma unroll
  for (int r = 0; r < 8; ++r) { minv[r] = 3.4e38f; mini[r] = 0; }

  const float* bp0 = &ldsW[0][col * LDS_STRIDE + koff];
  const float* bp1 = &ldsW[1][col * LDS_STRIDE + koff];

  // Prologue: async-fill buffer 0 with tile 0.
#pragma unroll
  for (int i = 0; i < 4; ++i)
    async_copy16(g0 + i * (4 * E_DIM), l0 + i * (4 * LDS_STRIDE));

  for (int tp = 0; tp < N_TILES; tp += 2) {
    // ---- stage A: prefetch tile tp+1 -> buf1, compute tile tp from buf0
    {
      const float* gt = g0 + (size_t)(tp + 1) * (TILE_N * E_DIM);
#pragma unroll
      for (int i = 0; i < 4; ++i)
        async_copy16(gt + i * (4 * E_DIM), l1 + i * (4 * LDS_STRIDE));
      wait_async_le4();   // in-order: oldest 4 (tile tp) have landed
      __syncthreads();    // all waves waited their portion -> tile visible

      v8f acc = {};
#pragma unroll
      for (int k = 0; k < 64; ++k) {
        v2f bf = *(const v2f*)(bp0 + 4 * k);
        acc = __builtin_amdgcn_wmma_f32_16x16x4_f32(
            false, A[k], false, bf, (short)0, acc, false, false);
      }
      const int   n  = tp * TILE_N + col;
      const float wn = wnorm[n];
#pragma unroll
      for (int r = 0; r < 8; ++r) {
        const float s = wn - 2.0f * acc[r];
        if (s < minv[r]) { minv[r] = s; mini[r] = n; }
      }
      __syncthreads();    // readers done before buf0 is re-targeted
    }
    // ---- stage B: prefetch tile (tp+2)&511 -> buf0, compute tile tp+1 from buf1
    {
      const int tn = (tp + 2) & (N_TILES - 1);  // wrap: dead prefetch on last pair
      const float* gt = g0 + (size_t)tn * (TILE_N * E_DIM);
#pragma unroll
      for (int i = 0; i < 4; ++i)
        async_copy16(gt + i * (4 * E_DIM), l0 + i * (4 * LDS_STRIDE));
      wait_async_le4();
      __syncthreads();

      v8f acc = {};
#pragma unroll
      for (int k = 0; k < 64; ++k) {
        v2f bf = *(const v2f*)(bp1 + 4 * k);
        acc = __builtin_amdgcn_wmma_f32_16x16x4_f32(
            false, A[k], false, bf, (short)0, acc, false, false);
      }
      const int   n  = (tp + 1) * TILE_N + col;
      const float wn = wnorm[n];
#pragma unroll
      for (int r = 0; r < 8; ++r) {
        const float s = wn - 2.0f * acc[r];
        if (s < minv[r]) { minv[r] = s; mini[r] = n; }
      }
      __syncthreads();
    }
  }

  // cross-lane argmin over the 16 lanes of each half-wave (C-layout: row r / r+8)
#pragma unroll
  for (int r = 0; r < 8; ++r) {
    float v  = minv[r];
    int   bi = mini[r];
#pragma unroll
    for (int mk = 8; mk >= 1; mk >>= 1) {
      const float ov = __shfl_xor(v, mk);
      const int   oi = __shfl_xor(bi, mk);
      if (ov < v || (ov == v && oi < bi)) { v = ov; bi = oi; }
    }
    if (col == 0) {
      const int row = rowBase + r + half * 8;
      idxI[row] = bi;
      idxF[row] = (float)bi;
      atomicAdd(&counts[bi], 1);
    }
  }
}

// ---------------- gather codebook, write transposed output, MSE loss ----------------
__global__ void vq_gather(const float* __restrict__ z, const float* __restrict__ w,
                          const int* __restrict__ idxI, float* __restrict__ zq,
                          float* __restrict__ sumsq) {
  __shared__ float red[256];
  const int t = threadIdx.x;
  const size_t stride = (size_t)gridDim.x * blockDim.x;
  float local = 0.0f;
  for (size_t o = (size_t)blockIdx.x * blockDim.x + t; o < (size_t)N_ROWS * E_DIM; o += stride) {
    const int bidx = (int)(o >> 20);        // E_DIM*LHW = 2^20
    const int c    = (int)(o >> 12) & 255;
    const int lhw  = (int)o & 4095;
    const int n    = (bidx << 12) | lhw;
    const float q  = w[(size_t)idxI[n] * E_DIM + c];
    zq[o] = q;                               // straight-through: z + sg(zq - z) == zq
    const float d = q - z[o];
    local += d * d;
  }
  red[t] = local;
  __syncthreads();
  for (int s = 128; s > 0; s >>= 1) { if (t < s) red[t] += red[t + s]; __syncthreads(); }
  if (t == 0) atomicAdd(sumsq, red[0]);
}

// ---------------- loss / perplexity / unique ----------------
__global__ void vq_finalize(const int* __restrict__ counts, const float* __restrict__ sumsq,
                            float* __restrict__ scalars) {
  __shared__ float redP[256];
  __shared__ int   redU[256];
  const int t = threadIdx.x;
  float pl = 0.0f;
  int   u  = 0;
  for (int i = t; i < N_E; i += 256) {
    const int   ci = counts[i];
    const float p  = (float)ci / (float)N_ROWS;
    pl += p * logf(p + 1e-10f);
    u  += (ci > 0) ? 1 : 0;
  }
  redP[t] = pl; redU[t] = u;
  __syncthreads();
  for (int s = 128; s > 0; s >>= 1) {
    if (t < s) { redP[t] += redP[t + s]; redU[t] += redU[t + s]; }
    __syncthreads();
  }
  if (t == 0) {
    scalars[0] = BETA * (*sumsq) / (float)((size_t)N_ROWS * E_DIM);  // loss
    scalars[1] = expf(-redP[0]);                                     // perplexity
    scalars[2] = (float)redU[0];                                     // unique
  }
}

extern "C" void kernel_launch(void* const* d_in, const int* in_sizes, int n_in,
                              void* d_out, int out_size, void* d_ws, size_t ws_size,
                              hipStream_t stream) {
  const float* z = (const float*)d_in[0];   // (4,256,16,16,16)
  const float* w = (const float*)d_in[1];   // (8192,256)
  float* out = (float*)d_out;

  char*  ws     = (char*)d_ws;
  float* sumsq  = (float*)ws;
  int*   counts = (int*)(ws + WS_COUNTS_OFF);
  int*   idxI   = (int*)(ws + WS_IDX_OFF);
  float* wnorm  = (float*)(ws + WS_WNORM_OFF);

  float* zq      = out;                                   // 4194304 floats
  float* scalars = out + (size_t)N_ROWS * E_DIM;          // loss, perplexity, unique
  float* idxF    = out + (size_t)N_ROWS * E_DIM + 3;      // 16384 indices (as float)

  vq_init    <<<32, 256, 0, stream>>>(sumsq, counts);
  vq_wnorm   <<<N_E / 8, 256, 0, stream>>>(w, wnorm);
  vq_argmin  <<<N_ROWS / 128, 256, 0, stream>>>(z, w, wnorm, idxI, idxF, counts);
  vq_gather  <<<4096, 256, 0, stream>>>(z, w, idxI, zq, sumsq);
  vq_finalize<<<1, 256, 0, stream>>>(counts, sumsq, scalars);
}